// FourierLayer_33191507263823
// MI455X (gfx1250) — compile-verified
//
#include <hip/hip_runtime.h>
#include <math.h>

typedef __attribute__((ext_vector_type(2))) float v2f;
typedef __attribute__((ext_vector_type(8))) float v8f;

#define B_SZ   16
#define T_LEN  4096
#define D_CH   256
#define FP     2048          // padded freq rows; freq index = row+1, row 2047 zeroed
#define M_ROWS (2 * FP)      // cos rows then -sin rows
#define K_TOP  7
#define TWO_PI 6.28318530717958647692f

// ---------------- kernel 1: DFT twiddle matrix ----------------
// A[m][t]: m<FP -> cos(2*pi*(m+1)*t/T); m>=FP -> -sin(2*pi*(m-FP+1)*t/T)
// so that  Re(X_f) = cos-rows @ x,  Im(X_f) = (-sin)-rows @ x.
// Nyquist pad rows (2047, 4095) are zeroed so they can never enter top-k.
__global__ __launch_bounds__(256) void build_twiddles(float* __restrict__ A) {
    int idx = blockIdx.x * 256 + threadIdx.x;     // < 4096*4096 = 16.7M
    int m = idx >> 12;                            // row
    int t = idx & (T_LEN - 1);                    // col
    int r = m & (FP - 1);
    float val = 0.0f;
    if (r != FP - 1) {
        int f = r + 1;
        int ph = (f * t) & (T_LEN - 1);           // exact: f*t < 2^24
        float ang = (float)ph * (TWO_PI / (float)T_LEN);
        val = (m < FP) ? __cosf(ang) : -__sinf(ang);
    }
    A[idx] = val;
}

// ---------------- kernel 2: DFT as f32 WMMA GEMM ----------------
// C[b] (4096 x 256) = A (4096 x 4096) @ x[b] (4096 x 256)
#define BM 128
#define BN 128
#define BK 16
#define SA 18     // LDS row stride for A tile (floats): 72B rows -> 8B aligned,
                  // 18*L mod 64 distinct for L=0..15 -> conflict-free frag loads
#define SB 132    // LDS row stride for B tile (floats)

__global__ __launch_bounds__(256) void dft_gemm(const float* __restrict__ A,
                                                const float* __restrict__ x,
                                                float* __restrict__ X) {
    __shared__ float As[BM * SA];
    __shared__ float Bs[BK * SB];

    const int tid  = threadIdx.x;
    const int m0   = blockIdx.x * BM;
    const int n0   = blockIdx.y * BN;
    const int b    = blockIdx.z;
    const int wave = tid >> 5;        // wave32
    const int lane = tid & 31;
    const int wm   = wave >> 1;       // 0..3  -> 32-row strip
    const int wn   = wave & 1;        // 0..1  -> 64-col strip
    const int lr   = lane & 15;       // row/col within 16x16 frag
    const int lh   = lane >> 4;       // k-half select (K+0/1 vs K+2/3)

    v8f acc[2][4];
#pragma unroll
    for (int i = 0; i < 2; ++i)
#pragma unroll
        for (int j = 0; j < 4; ++j) { v8f z = {}; acc[i][j] = z; }

    const float* xb = x + (size_t)b * T_LEN * D_CH;

    for (int k0 = 0; k0 < T_LEN; k0 += BK) {
        // stage A tile (128 x 16): 8 floats per thread
        {
            int r = tid >> 1, h = (tid & 1) * 8;
            const float* src = A + (size_t)(m0 + r) * T_LEN + k0 + h;
            float* dst = &As[r * SA + h];
#pragma unroll
            for (int j = 0; j < 8; ++j) dst[j] = src[j];
        }
        // stage B tile (16 x 128): 8 floats per thread
        {
            int r = tid >> 4, c = (tid & 15) * 8;
            const float* src = xb + (size_t)(k0 + r) * D_CH + n0 + c;
            float* dst = &Bs[r * SB + c];
#pragma unroll
            for (int j = 0; j < 8; ++j) dst[j] = src[j];
        }
        __syncthreads();

#pragma unroll
        for (int kk = 0; kk < BK; kk += 4) {
            const int ko = kk + 2 * lh;           // K offset per lane-half
            v2f af[2], bf[4];
#pragma unroll
            for (int i = 0; i < 2; ++i) {         // A frag: lanes hold rows, 2 K's
                const float* p = &As[(wm * 32 + i * 16 + lr) * SA + ko];
                v2f a; a[0] = p[0]; a[1] = p[1];
                af[i] = a;
            }
#pragma unroll
            for (int j = 0; j < 4; ++j) {         // B frag: lanes hold cols, 2 K's
                const int col = wn * 64 + j * 16 + lr;
                v2f bb; bb[0] = Bs[ko * SB + col]; bb[1] = Bs[(ko + 1) * SB + col];
                bf[j] = bb;
            }
#pragma unroll
            for (int i = 0; i < 2; ++i)
#pragma unroll
                for (int j = 0; j < 4; ++j)
                    acc[i][j] = __builtin_amdgcn_wmma_f32_16x16x4_f32(
                        false, af[i], false, bf[j], (short)0, acc[i][j],
                        false, false);
        }
        __syncthreads();
    }

    // store C: VGPR v holds M=v (lanes 0-15) / M=v+8 (lanes 16-31), N=lane&15
    float* Xb = X + (size_t)b * M_ROWS * D_CH;
#pragma unroll
    for (int i = 0; i < 2; ++i)
#pragma unroll
        for (int j = 0; j < 4; ++j) {
            const int gcol = n0 + wn * 64 + j * 16 + lr;
#pragma unroll
            for (int v = 0; v < 8; ++v) {
                const int grow = m0 + wm * 32 + i * 16 + v + 8 * lh;
                Xb[(size_t)grow * D_CH + gcol] = acc[i][j][v];
            }
        }
}

// ---------------- kernel 3: per-channel top-7 by |X|^2 ----------------
__global__ __launch_bounds__(256) void topk_select(const float* __restrict__ X,
                                                   float* __restrict__ ampO,
                                                   float* __restrict__ phO,
                                                   int* __restrict__ fqO) {
    const int d = threadIdx.x;
    const int b = blockIdx.x;
    const float* Xb = X + (size_t)b * M_ROWS * D_CH;

    float bv[K_TOP];
    int   bi[K_TOP];
#pragma unroll
    for (int k = 0; k < K_TOP; ++k) { bv[k] = -1.0f; bi[k] = 0; }

    for (int m = 0; m < FP - 1; ++m) {            // freq bins 1..2047
        const float re = Xb[(size_t)m * D_CH + d];
        const float im = Xb[(size_t)(m + FP) * D_CH + d];
        const float m2 = re * re + im * im;
        if (m2 > bv[0]) {                         // replace min, bubble once
            bv[0] = m2; bi[0] = m;
#pragma unroll
            for (int j = 0; j < K_TOP - 1; ++j) {
                if (bv[j] > bv[j + 1]) {
                    float tv = bv[j]; bv[j] = bv[j + 1]; bv[j + 1] = tv;
                    int   ti = bi[j]; bi[j] = bi[j + 1]; bi[j + 1] = ti;
                }
            }
        }
    }

    const size_t base = ((size_t)b * D_CH + d) * K_TOP;
#pragma unroll
    for (int k = 0; k < K_TOP; ++k) {
        const int m = bi[k];
        const float re = Xb[(size_t)m * D_CH + d];
        const float im = Xb[(size_t)(m + FP) * D_CH + d];
        ampO[base + k] = (4.0f / (float)T_LEN) * sqrtf(fmaxf(bv[k], 0.0f));
        phO[base + k]  = atan2f(im, re);
        fqO[base + k]  = m + 1;
    }
}

// ---------------- kernel 4: synthesis ----------------
#define TCHUNK 32
__global__ __launch_bounds__(256) void synth(const float* __restrict__ ampO,
                                             const float* __restrict__ phO,
                                             const int* __restrict__ fqO,
                                             float* __restrict__ out) {
    const int d  = threadIdx.x;
    const int b  = blockIdx.y;
    const int t0 = blockIdx.x * TCHUNK;
    const size_t base = ((size_t)b * D_CH + d) * K_TOP;

    float amp[K_TOP], ph[K_TOP];
    int   fq[K_TOP];
#pragma unroll
    for (int k = 0; k < K_TOP; ++k) {
        amp[k] = ampO[base + k];
        ph[k]  = phO[base + k];
        fq[k]  = fqO[base + k];
    }

    float* ob = out + ((size_t)b * T_LEN + t0) * D_CH + d;
    for (int tt = 0; tt < TCHUNK; ++tt) {
        const int t = t0 + tt;
        float s = 0.0f;
#pragma unroll
        for (int k = 0; k < K_TOP; ++k) {
            const int r = (fq[k] * t) & (T_LEN - 1);    // exact phase reduction
            const float ang = (float)r * (TWO_PI / (float)T_LEN) + ph[k];
            s += amp[k] * __cosf(ang);
        }
        ob[(size_t)tt * D_CH] = s;
    }
}

// ---------------- launcher ----------------
extern "C" void kernel_launch(void* const* d_in, const int* in_sizes, int n_in,
                              void* d_out, int out_size, void* d_ws, size_t ws_size,
                              hipStream_t stream) {
    const float* x = (const float*)d_in[0];
    float* out = (float*)d_out;
    char* ws = (char*)d_ws;

    // workspace layout (bytes):
    //   [0, 64MB)    : twiddle matrix A (4096 x 4096 f32)
    //   [64, 128MB)  : spectrum X (16 x 4096 x 256 f32)  [cos rows; -sin rows]
    //   [128MB, ...) : selection: amp, phase (f32), freq (i32), 4096*7 each
    float* Atw  = (float*)ws;
    float* Xall = (float*)(ws + (size_t)M_ROWS * T_LEN * sizeof(float));
    size_t selOff = (size_t)M_ROWS * T_LEN * sizeof(float)
                  + (size_t)B_SZ * M_ROWS * D_CH * sizeof(float);
    float* ampO = (float*)(ws + selOff);
    float* phO  = ampO + (size_t)B_SZ * D_CH * K_TOP;
    int*   fqO  = (int*)(phO + (size_t)B_SZ * D_CH * K_TOP);

    build_twiddles<<<(M_ROWS * T_LEN) / 256, 256, 0, stream>>>(Atw);

    dim3 gg(M_ROWS / BM, D_CH / BN, B_SZ);
    dft_gemm<<<gg, 256, 0, stream>>>(Atw, x, Xall);

    topk_select<<<B_SZ, D_CH, 0, stream>>>(Xall, ampO, phO, fqO);

    dim3 gs(T_LEN / TCHUNK, B_SZ);
    synth<<<gs, D_CH, 0, stream>>>(ampO, phO, fqO, out);
}